// ONE_39436389712021
// MI455X (gfx1250) — compile-verified
//
#include <hip/hip_runtime.h>

// Problem constants (from reference)
#define N_SENT 65536
#define D_FEAT 2304
#define N_BAG  4096
#define N_CLS  53
#define CPAD   64        // classes padded to 4 WMMA tiles of 16
#define CHUNK  128       // K-chunk of W staged in LDS: 64*128*4 = 32 KB

typedef __attribute__((ext_vector_type(2))) float v2f;
typedef __attribute__((ext_vector_type(8))) float v8f;

// ---------------------------------------------------------------------------
// Pass 1: logits = reps*W^T + b, softmax -> probs [N, 53] in workspace.
// One wave = 16 sentences x 64 classes, V_WMMA_F32_16X16X4_F32 over K=2304.
// Block = 256 threads = 8 waves = 128 sentences. Grid = 512 blocks.
// ---------------------------------------------------------------------------
__global__ __launch_bounds__(256) void dense_softmax_kernel(
    const float* __restrict__ reps, const float* __restrict__ W,
    const float* __restrict__ bias, float* __restrict__ probs)
{
  __shared__ float Ws[CPAD * CHUNK];   // Ws[c][k] = W[c][k0+k], zero-padded c>=53

  const int tid  = threadIdx.x;
  const int wave = tid >> 5;
  const int lane = tid & 31;
  const int half = lane >> 4;          // which 16-lane half of the wave
  const int lid  = lane & 15;
  const int srow = blockIdx.x * 128 + wave * 16;   // first sentence of tile

  v8f acc[4] = {v8f{}, v8f{}, v8f{}, v8f{}};

  for (int k0 = 0; k0 < D_FEAT; k0 += CHUNK) {
    // cooperative stage of the W chunk (64 padded classes x CHUNK)
    #pragma unroll
    for (int i = 0; i < (CPAD * CHUNK) / 256; ++i) {
      int e = tid + i * 256;
      int c = e / CHUNK;
      int k = e % CHUNK;
      Ws[e] = (c < N_CLS) ? W[(size_t)c * D_FEAT + k0 + k] : 0.0f;
    }
    __syncthreads();

    // A-fragment base: lane l -> row (srow + l%16), k = k0 + kk + 2*(l/16)
    const float* arow = reps + (size_t)(srow + lid) * D_FEAT + k0 + (half << 1);

    #pragma unroll 8
    for (int kk = 0; kk < CHUNK; kk += 4) {
      v2f a;
      a.x = arow[kk];
      a.y = arow[kk + 1];
      const int koff = kk + (half << 1);
      #pragma unroll
      for (int t = 0; t < 4; ++t) {
        // B-fragment (mirrors A layout): lane l -> col (t*16 + l%16), same k
        const float* bp = &Ws[(t * 16 + lid) * CHUNK + koff];
        v2f bf;
        bf.x = bp[0];
        bf.y = bp[1];
        // 8 args: (neg_a, A, neg_b, B, c_mod, C, reuse_a, reuse_b)
        acc[t] = __builtin_amdgcn_wmma_f32_16x16x4_f32(
            false, a, false, bf, (short)0, acc[t], false, false);
      }
    }
    __syncthreads();
  }

  // Bias (+ mask for padded classes 53..63 so exp underflows to zero)
  float bias4[4];
  #pragma unroll
  for (int t = 0; t < 4; ++t) {
    int c = t * 16 + lid;
    bias4[t] = (c < N_CLS) ? bias[c] : -1.0e30f;
  }

  // C/D layout: lane l, vgpr v  ->  row M = v + 8*(l/16), col N = l%16
  #pragma unroll
  for (int v = 0; v < 8; ++v) {
    float x0 = acc[0][v] + bias4[0];
    float x1 = acc[1][v] + bias4[1];
    float x2 = acc[2][v] + bias4[2];
    float x3 = acc[3][v] + bias4[3];

    float m = fmaxf(fmaxf(x0, x1), fmaxf(x2, x3));
    #pragma unroll
    for (int msk = 1; msk <= 8; msk <<= 1)
      m = fmaxf(m, __shfl_xor(m, msk, 32));   // reduce within 16-lane half

    float e0 = __expf(x0 - m);
    float e1 = __expf(x1 - m);
    float e2 = __expf(x2 - m);
    float e3 = __expf(x3 - m);
    float s = e0 + e1 + e2 + e3;
    #pragma unroll
    for (int msk = 1; msk <= 8; msk <<= 1)
      s += __shfl_xor(s, msk, 32);
    float inv = 1.0f / s;

    int j = srow + v + (half << 3);
    float* prow = probs + (size_t)j * N_CLS;
    float ev[4] = {e0, e1, e2, e3};
    #pragma unroll
    for (int t = 0; t < 4; ++t) {
      int c = t * 16 + lid;
      if (c < N_CLS) prow[c] = ev[t] * inv;
    }
  }
}

// ---------------------------------------------------------------------------
// Pass 2: lp[j] = probs[j, label[bag(j)]], bag via binary search on scope.
// ---------------------------------------------------------------------------
__global__ __launch_bounds__(256) void lp_kernel(
    const float* __restrict__ probs, const int* __restrict__ scope,
    const int* __restrict__ label, float* __restrict__ lp)
{
  int j = blockIdx.x * blockDim.x + threadIdx.x;
  if (j >= N_SENT) return;
  // largest i with scope[i] <= j  (scope[0]=0, scope[B]=N)
  int lo = 0, hi = N_BAG;
  while (hi - lo > 1) {
    int mid = (lo + hi) >> 1;
    if (scope[mid] <= j) lo = mid; else hi = mid;
  }
  lp[j] = probs[(size_t)j * N_CLS + label[lo]];
}

// ---------------------------------------------------------------------------
// Pass 3: per bag, first index achieving max lp; copy its prob row to out.
// One wave32 per bag.
// ---------------------------------------------------------------------------
__global__ __launch_bounds__(32) void select_kernel(
    const float* __restrict__ probs, const int* __restrict__ scope,
    const float* __restrict__ lp, float* __restrict__ out)
{
  int bag  = blockIdx.x;
  int lane = threadIdx.x;
  int s = scope[bag];
  int e = scope[bag + 1];

  float bestv = -1.0e30f;
  int   besti = N_SENT;
  for (int j = s + lane; j < e; j += 32) {   // per-lane j strictly increasing
    float v = lp[j];
    if (v > bestv) { bestv = v; besti = j; } // keeps FIRST max within lane
  }
  // cross-lane reduce: larger value wins; ties -> smaller index (first max)
  #pragma unroll
  for (int msk = 16; msk; msk >>= 1) {
    float ov = __shfl_xor(bestv, msk, 32);
    int   oi = __shfl_xor(besti, msk, 32);
    if (ov > bestv || (ov == bestv && oi < besti)) { bestv = ov; besti = oi; }
  }
  int sel = besti < N_SENT ? besti : N_SENT - 1;   // clip (bags never empty)

  const float* prow = probs + (size_t)sel * N_CLS;
  for (int c = lane; c < N_CLS; c += 32)
    out[(size_t)bag * N_CLS + c] = prow[c];
}

// ---------------------------------------------------------------------------
extern "C" void kernel_launch(void* const* d_in, const int* in_sizes, int n_in,
                              void* d_out, int out_size, void* d_ws, size_t ws_size,
                              hipStream_t stream)
{
  const float* reps  = (const float*)d_in[0];   // [N, D]
  const int*   scope = (const int*)  d_in[1];   // [B+1]
  const int*   label = (const int*)  d_in[2];   // [B]
  const float* W     = (const float*)d_in[3];   // [C, D]
  const float* bias  = (const float*)d_in[4];   // [C]
  float* out = (float*)d_out;                   // [B, C]

  float* probs = (float*)d_ws;                          // N*C floats (~13.3 MiB)
  float* lp    = probs + (size_t)N_SENT * N_CLS;        // N floats

  dense_softmax_kernel<<<N_SENT / 128, 256, 0, stream>>>(reps, W, bias, probs);
  lp_kernel<<<N_SENT / 256, 256, 0, stream>>>(probs, scope, label, lp);
  select_kernel<<<N_BAG, 32, 0, stream>>>(probs, scope, lp, out);
}